// upinterp1d_22548578304374
// MI455X (gfx1250) — compile-verified
//
#include <hip/hip_runtime.h>
#include <hip/hip_bf16.h>

// 2x cubic (Keys a=-0.5) upsampling along length of (8, L, 16) f32, channels-last.
// Per 16x16 output tile: Out[16 x 16] = W[16 x 12] * X[12 x 16], computed with
// three V_WMMA_F32_16X16X4_F32 accumulations per wave32.
// Split: interior kernel (uniform scalar control flow, no edge logic) +
// tiny edge kernel (first/last tile of each batch, renormalized weights).

typedef __attribute__((ext_vector_type(2))) float v2f;
typedef __attribute__((ext_vector_type(8))) float v8f;

#define WAVES_PER_BLOCK 8
#define TILES_PER_WAVE  8
#define TILES_PER_BLOCK (WAVES_PER_BLOCK * TILES_PER_WAVE)

// Keys cubic a=-0.5, upsample-by-2, half-pixel centers.
// Even output rows: phase 0.75 -> (-3, 29, 111, -9)/128 over taps i-2..i+1.
// Odd output rows:  phase 0.25 -> reversed, over taps i-1..i+2.
__device__ __forceinline__ float tap_w(int m, int t) {
    int tt = (m & 1) ? (3 - t) : t;
    float r = (tt == 0) ? -0.0234375f
            : (tt == 1) ?  0.2265625f
            : (tt == 2) ?  0.8671875f
            :              -0.0703125f;
    return r;
}

// Interior tile weight matrix W[m][k]; k relative to j0 = i0-2 (0..11).
__device__ __forceinline__ float tile_w_int(int m, int k) {
    int off = k - ((m >> 1) + (m & 1));
    if (off < 0 || off > 3) return 0.0f;
    return tap_w(m, off);
}

// Boundary-aware W[m][k]: drop taps outside [0, L), renormalize the row by the
// surviving weight sum (jax.image.resize semantics). Branchless (selects only).
__device__ __forceinline__ float tile_w_edge(int m, int k, long j0, long L) {
    int base = (m >> 1) + (m & 1);
    float wsum = 0.0f;
#pragma unroll
    for (int t = 0; t < 4; ++t) {
        long j = j0 + base + t;
        wsum += (j >= 0 && j < L) ? tap_w(m, t) : 0.0f;
    }
    int off = k - base;
    long jk = j0 + k;
    bool valid = (off >= 0) & (off <= 3) & (jk >= 0) & (jk < L);
    float w = tap_w(m, off & 3);
    return valid ? (w / wsum) : 0.0f;
}

__device__ __forceinline__ long clamp_row(long r, long L) {
    return r < 0 ? 0 : (r >= L ? L - 1 : r);
}

// ---------------- Interior kernel: tiles [1, tilesPerBatch-2] ----------------
__global__ __launch_bounds__(256) void upcubic2x_interior_wmma(
    const float* __restrict__ x, float* __restrict__ y,
    int L, int tilesPerBatch) {
    const int lane = threadIdx.x & 31;
    // Force wave id into an SGPR so all per-tile control flow is scalar and
    // EXEC stays all-ones (WMMA requirement).
    const int wave = __builtin_amdgcn_readfirstlane((int)(threadIdx.x >> 5));
    const int half = lane >> 4;   // K half: lanes 16-31 hold K = +2,+3 per step
    const int m    = lane & 15;   // A row (output row within tile)
    const int c    = lane & 15;   // B/D column (channel)
    const int batch = blockIdx.y;

    // Interior A registers: step s consumes K = 4s + 2*half + {0,1}.
    v2f A0, A1, A2;
    A0[0] = tile_w_int(m, 0 + 2 * half + 0);
    A0[1] = tile_w_int(m, 0 + 2 * half + 1);
    A1[0] = tile_w_int(m, 4 + 2 * half + 0);
    A1[1] = tile_w_int(m, 4 + 2 * half + 1);
    A2[0] = tile_w_int(m, 8 + 2 * half + 0);
    A2[1] = tile_w_int(m, 8 + 2 * half + 1);

    const long Lz = (long)L;
    // Lane-fixed components folded into base pointers once:
    //  input row j0+2*half, channel c; output row o0+8*half, channel c.
    const float* xb = x + (long)batch * Lz * 16 + 32 * half + c - 32;  // -32: j0 = tile*8 - 2
    float*       yb = y + (long)batch * (2 * Lz) * 16 + 128 * half + c;

    const int tile0 = 1 + blockIdx.x * TILES_PER_BLOCK + wave;  // skip edge tile 0
    const int lastInterior = tilesPerBatch - 2;

#pragma unroll
    for (int t = 0; t < TILES_PER_WAVE; ++t) {
        const int tile = tile0 + t * WAVES_PER_BLOCK;  // scalar
        if (tile > lastInterior) return;               // scalar branch; monotonic

        const float* p = xb + (long)tile * 128;  // tile*8 input rows * 16 ch
        v2f B0, B1, B2;
        B0[0] = p[0];    // row j0+2h+0
        B0[1] = p[16];   // row j0+2h+1
        B1[0] = p[64];   // row j0+2h+4
        B1[1] = p[80];   // row j0+2h+5
        B2[0] = p[128];  // row j0+2h+8
        B2[1] = p[144];  // row j0+2h+9

        v8f acc = {0.f, 0.f, 0.f, 0.f, 0.f, 0.f, 0.f, 0.f};
        acc = __builtin_amdgcn_wmma_f32_16x16x4_f32(false, A0, false, B0,
                                                    (short)0, acc, false, false);
        acc = __builtin_amdgcn_wmma_f32_16x16x4_f32(false, A1, false, B1,
                                                    (short)0, acc, false, false);
        acc = __builtin_amdgcn_wmma_f32_16x16x4_f32(false, A2, false, B2,
                                                    (short)0, acc, false, false);

        float* q = yb + (long)tile * 256;  // tile*16 output rows * 16 ch
#pragma unroll
        for (int v = 0; v < 8; ++v) q[v * 16] = acc[v];
    }
}

// ---------------- Edge kernel: tile 0 and tile tilesPerBatch-1 per batch -----
// One block of 16 waves: wave w -> batch w/2, first/last tile by w&1.
__global__ __launch_bounds__(512) void upcubic2x_edge_wmma(
    const float* __restrict__ x, float* __restrict__ y,
    int L, int tilesPerBatch) {
    const int lane = threadIdx.x & 31;
    const int wave = __builtin_amdgcn_readfirstlane((int)(threadIdx.x >> 5));
    const int half = lane >> 4;
    const int m    = lane & 15;
    const int c    = lane & 15;

    const int batch  = wave >> 1;
    const int tile   = (wave & 1) ? (tilesPerBatch - 1) : 0;

    const long Lz = (long)L;
    const long j0 = (long)tile * 8 - 2;
    const float* xb = x + (long)batch * Lz * 16 + c;
    float*       yb = y + (long)batch * (2 * Lz) * 16 + c;

    v2f A0, A1, A2;
    A0[0] = tile_w_edge(m, 0 + 2 * half + 0, j0, Lz);
    A0[1] = tile_w_edge(m, 0 + 2 * half + 1, j0, Lz);
    A1[0] = tile_w_edge(m, 4 + 2 * half + 0, j0, Lz);
    A1[1] = tile_w_edge(m, 4 + 2 * half + 1, j0, Lz);
    A2[0] = tile_w_edge(m, 8 + 2 * half + 0, j0, Lz);
    A2[1] = tile_w_edge(m, 8 + 2 * half + 1, j0, Lz);

    const long r = j0 + 2 * half;
    v2f B0, B1, B2;
    B0[0] = xb[clamp_row(r + 0, Lz) * 16];
    B0[1] = xb[clamp_row(r + 1, Lz) * 16];
    B1[0] = xb[clamp_row(r + 4, Lz) * 16];
    B1[1] = xb[clamp_row(r + 5, Lz) * 16];
    B2[0] = xb[clamp_row(r + 8, Lz) * 16];
    B2[1] = xb[clamp_row(r + 9, Lz) * 16];

    v8f acc = {0.f, 0.f, 0.f, 0.f, 0.f, 0.f, 0.f, 0.f};
    acc = __builtin_amdgcn_wmma_f32_16x16x4_f32(false, A0, false, B0,
                                                (short)0, acc, false, false);
    acc = __builtin_amdgcn_wmma_f32_16x16x4_f32(false, A1, false, B1,
                                                (short)0, acc, false, false);
    acc = __builtin_amdgcn_wmma_f32_16x16x4_f32(false, A2, false, B2,
                                                (short)0, acc, false, false);

    float* q = yb + ((long)tile * 16 + 8 * half) * 16;
#pragma unroll
    for (int v = 0; v < 8; ++v) q[v * 16] = acc[v];
}

extern "C" void kernel_launch(void* const* d_in, const int* in_sizes, int n_in,
                              void* d_out, int out_size, void* d_ws, size_t ws_size,
                              hipStream_t stream) {
    const float* x = (const float*)d_in[0];
    float* y = (float*)d_out;

    const int B = 8, C = 16;                 // per reference setup_inputs()
    const int L = in_sizes[0] / (B * C);     // 262144
    const int tilesPerBatch = (2 * L) / 16;  // 16 output rows per tile

    const int interiorTiles = tilesPerBatch - 2;
    const int blocksPerBatch =
        (interiorTiles + TILES_PER_BLOCK - 1) / TILES_PER_BLOCK;

    dim3 gridI(blocksPerBatch, B, 1);
    upcubic2x_interior_wmma<<<gridI, 256, 0, stream>>>(x, y, L, tilesPerBatch);
    upcubic2x_edge_wmma<<<dim3(1, 1, 1), 2 * B * 32, 0, stream>>>(x, y, L,
                                                                  tilesPerBatch);
}